// FasterDMFNSMultiHeadAttention_54408645706198
// MI455X (gfx1250) — compile-verified
//
#include <hip/hip_runtime.h>

// ---------------------------------------------------------------------------
// FasterDMFNS multi-head attention, fused two-pass, f32 WMMA + TDM (gfx1250)
// ---------------------------------------------------------------------------

typedef float        v2f  __attribute__((ext_vector_type(2)));
typedef float        v8f  __attribute__((ext_vector_type(8)));
typedef unsigned int u32x4 __attribute__((ext_vector_type(4)));
typedef int          i32x4 __attribute__((ext_vector_type(4)));
typedef int          i32x8 __attribute__((ext_vector_type(8)));

#define WMMA_F32(a, b, c) \
    __builtin_amdgcn_wmma_f32_16x16x4_f32(false, (a), false, (b), (short)0, (c), false, false)

#if defined(__has_builtin)
#if __has_builtin(__builtin_amdgcn_tensor_load_to_lds) && \
    __has_builtin(__builtin_amdgcn_s_wait_tensorcnt)
#define USE_TDM 1
#endif
#endif
#ifndef USE_TDM
#define USE_TDM 0
#endif

constexpr int Bb  = 4;
constexpr int Ss  = 2048;
constexpr int HID = 768;
constexpr int Hh  = 12;
constexpr int Dd  = 64;
constexpr int TRIPLE = 3 * HID;            // 2304, qkv row stride
constexpr float PPOW = -(Dd + 1.5f);       // -(D + alpha) = -65.5
constexpr float EPSF = 1e-12f;

#if USE_TDM
// ---------------------------------------------------------------------------
// Tensor Data Mover: load a (tile_d2 x 16 x 64) f32 tile (innermost contiguous
// 64 elems, 16 rows of stride 2304 elems, optional 2 slices of stride
// `stride1` elems) from global into contiguous LDS at byte offset lds_off.
// Descriptor packing per CDNA5 ISA (D# groups 0..3).
// This toolchain's builtin takes 6 args: (g0, g1, g2, g3, extra, cpol);
// the extra 8-dword group is passed zero-filled.
// ---------------------------------------------------------------------------
__device__ __forceinline__ void tdm_tile_load(unsigned lds_off, const float* gptr,
                                              unsigned tile_d2, unsigned stride1)
{
    unsigned long long ga = (unsigned long long)(size_t)gptr;
    u32x4 g0;
    g0.x = 1u;                                             // count=1, user D#
    g0.y = lds_off;                                        // lds_addr [63:32]
    g0.z = (unsigned)ga;                                   // global_addr lo
    g0.w = (unsigned)((ga >> 32) & 0x01FFFFFFull) | (2u << 30);  // addr hi | type=2
    i32x8 g1;
    g1[0] = (int)(2u << 16);                // workgroup_mask=0, data_size=2 (4B)
    g1[1] = (int)(64u << 16);               // tensor_dim0 = 64   (bits 63:48)
    g1[2] = (int)(16u << 16);               // tensor_dim1 = 16   (bits 95:80)
    g1[3] = (int)(64u << 16);               // tile_dim0   = 64   (bits 127:112)
    g1[4] = (int)(16u | (tile_d2 << 16));   // tile_dim1=16, tile_dim2 (159:144)
    g1[5] = (int)(unsigned)TRIPLE;          // tensor_dim0_stride lo32 = 2304
    g1[6] = (int)(stride1 << 16);           // dim0_stride hi=0 | dim1_stride lo16
    g1[7] = 0;                              // dim1_stride hi
    i32x4 g2;
    g2[0] = (int)tile_d2;                   // tensor_dim2 (0 => 2D)
    g2[1] = 0; g2[2] = 0; g2[3] = 0;
    i32x4 g3 = {0, 0, 0, 0};
    i32x8 g4 = {0, 0, 0, 0, 0, 0, 0, 0};    // extra group (6-arg toolchain form)
    __builtin_amdgcn_tensor_load_to_lds(g0, g1, g2, g3, g4, 0);
}
#define TDM_WAIT() do { __builtin_amdgcn_s_wait_tensorcnt(0); asm volatile("" ::: "memory"); } while (0)
#endif

// ---------------------------------------------------------------------------
// Generic f32 WMMA GEMM: C[M,N] = A[M,K] @ W[K,N] + bias[N]
// One wave -> one 16(M) x 64(N) tile; K stepped by 4; register double-buffered
// so the next step's fragments load while the current WMMAs issue.
// ---------------------------------------------------------------------------
__global__ __launch_bounds__(256)
void gemm_wmma_f32(const float* __restrict__ A, const float* __restrict__ W,
                   const float* __restrict__ bias, float* __restrict__ C,
                   int M, int N, int K)
{
    const int lane = threadIdx.x & 31;
    const int wave = threadIdx.x >> 5;
    const int half = lane >> 4;
    const int lm   = lane & 15;

    const int ntN   = N >> 6;
    const int tiles = (M >> 4) * ntN;
    const int t = blockIdx.x * 8 + wave;          // wave-uniform
    if (t >= tiles) return;
    const int mt = t / ntN, nt = t % ntN;
    const int m0 = mt << 4, n0 = nt << 6;

    v8f acc0 = {}, acc1 = {}, acc2 = {}, acc3 = {};
    const float* ap = A + (size_t)(m0 + lm) * K + 2 * half;
    const float* wp = W + (size_t)(2 * half) * N + n0 + lm;
    const size_t wstep = (size_t)4 * N;

    // prologue fragment loads (k = 0)
    v2f a = *(const v2f*)ap;
    v2f b0, b1, b2, b3;
    b0.x = wp[ 0]; b0.y = wp[(size_t)N +  0];
    b1.x = wp[16]; b1.y = wp[(size_t)N + 16];
    b2.x = wp[32]; b2.y = wp[(size_t)N + 32];
    b3.x = wp[48]; b3.y = wp[(size_t)N + 48];

    for (int k = 0; k < K; k += 4) {
        const float* apn = (k + 4 < K) ? (ap + 4)   : ap;
        const float* wpn = (k + 4 < K) ? (wp + wstep) : wp;
        // issue next-step loads before consuming current fragments
        v2f an = *(const v2f*)apn;
        v2f c0, c1, c2, c3;
        c0.x = wpn[ 0]; c0.y = wpn[(size_t)N +  0];
        c1.x = wpn[16]; c1.y = wpn[(size_t)N + 16];
        c2.x = wpn[32]; c2.y = wpn[(size_t)N + 32];
        c3.x = wpn[48]; c3.y = wpn[(size_t)N + 48];

        acc0 = WMMA_F32(a, b0, acc0);
        acc1 = WMMA_F32(a, b1, acc1);
        acc2 = WMMA_F32(a, b2, acc2);
        acc3 = WMMA_F32(a, b3, acc3);

        a = an; b0 = c0; b1 = c1; b2 = c2; b3 = c3;
        ap = apn; wp = wpn;
    }

    const float bi0 = bias[n0 +  0 + lm];
    const float bi1 = bias[n0 + 16 + lm];
    const float bi2 = bias[n0 + 32 + lm];
    const float bi3 = bias[n0 + 48 + lm];
#pragma unroll
    for (int r = 0; r < 8; ++r) {
        float* crow = C + (size_t)(m0 + r + 8 * half) * N + n0 + lm;
        crow[ 0] = acc0[r] + bi0;
        crow[16] = acc1[r] + bi1;
        crow[32] = acc2[r] + bi2;
        crow[48] = acc3[r] + bi3;
    }
}

// ---------------------------------------------------------------------------
// Per-(b,h,s) squared norms of q and k rows.
// ---------------------------------------------------------------------------
__global__ __launch_bounds__(256)
void norms_kernel(const float* __restrict__ qkv,
                  float* __restrict__ qn, float* __restrict__ kn)
{
    const int idx = blockIdx.x * 256 + threadIdx.x;        // over B*H*S
    if (idx >= Bb * Hh * Ss) return;
    const int s  = idx % Ss;
    const int bh = idx / Ss;
    const int b  = bh / Hh, h = bh % Hh;

    const float* qp = qkv + (size_t)(b * Ss + s) * TRIPLE + h * Dd;
    const float* kp = qp + HID;
    float sq = 0.f, sk = 0.f;
#pragma unroll
    for (int d = 0; d < Dd; d += 4) {
        float4 q4 = *(const float4*)(qp + d);
        float4 k4 = *(const float4*)(kp + d);
        sq += q4.x * q4.x + q4.y * q4.y + q4.z * q4.z + q4.w * q4.w;
        sk += k4.x * k4.x + k4.y * k4.y + k4.z * k4.z + k4.w * k4.w;
    }
    qn[idx] = sq;
    kn[idx] = sk;
}

// ---------------------------------------------------------------------------
// Pass 1: column sums N_C[bh][t] = sum_s (1 + d(q_s,k_t))^PPOW.
// One wave per (bh, 16-wide t-tile); K fragments resident in registers,
// Q tiles streamed via TDM into double-buffered LDS.
// ---------------------------------------------------------------------------
__global__ __launch_bounds__(32)
void colsum_kernel(const float* __restrict__ qkv, const float* __restrict__ qn,
                   const float* __restrict__ kn, float* __restrict__ colsum)
{
#if USE_TDM
    __shared__ float qtile[2][16 * 64];     // 2 x 4KB double buffer
#endif
    const int lane = threadIdx.x;
    const int half = lane >> 4;
    const int lm   = lane & 15;
    const int widx = blockIdx.x;
    const int bh = widx >> 7;               // / (S/16)
    const int tt = widx & 127;
    const int b  = bh / Hh, h = bh % Hh;
    const int t0 = tt << 4;

    const float* qbase = qkv + (size_t)b * Ss * TRIPLE + h * Dd;
    const float* kbase = qbase + HID;

    // resident B fragments of K^T (d x t): 16 chunks of K=4
    v2f kb[16];
#pragma unroll
    for (int c = 0; c < 16; ++c)
        kb[c] = *(const v2f*)(kbase + (size_t)(t0 + lm) * TRIPLE + 4 * c + 2 * half);

    const float knv = kn[(size_t)bh * Ss + t0 + lm];
    float colacc = 0.f;

#if USE_TDM
    const unsigned q_off = (unsigned)(size_t)&qtile[0][0];
    tdm_tile_load(q_off, qbase, 0, 0);                       // prologue: s0 = 0
#endif

    for (int s0 = 0; s0 < Ss; s0 += 16) {
        v8f dot = {};
#if USE_TDM
        const int cur = (s0 >> 4) & 1;
        TDM_WAIT();
        if (s0 + 16 < Ss)
            tdm_tile_load(q_off + (unsigned)((cur ^ 1) * 4096),
                          qbase + (size_t)(s0 + 16) * TRIPLE, 0, 0);
        const float* qt = &qtile[cur][0];
#pragma unroll
        for (int c = 0; c < 16; ++c) {
            v2f a = *(const v2f*)(qt + lm * 64 + 4 * c + 2 * half);
            dot = WMMA_F32(a, kb[c], dot);
        }
#else
        if (s0 + 16 < Ss)
            __builtin_prefetch(qbase + (size_t)(s0 + 16 + lm) * TRIPLE, 0, 0);
#pragma unroll
        for (int c = 0; c < 16; ++c) {
            v2f a = *(const v2f*)(qbase + (size_t)(s0 + lm) * TRIPLE + 4 * c + 2 * half);
            dot = WMMA_F32(a, kb[c], dot);
        }
#endif
        const float qv = qn[(size_t)bh * Ss + s0 + lm];
#pragma unroll
        for (int r = 0; r < 8; ++r) {
            float qrow = __shfl(qv, r + 8 * half, 32);       // qn[s0 + r + 8*half]
            float d2 = qrow + knv - 2.0f * dot[r];
            float g  = sqrtf(fmaxf(d2, 0.0f));
            colacc  += __powf(1.0f + g, PPOW);
        }
    }
    colacc += __shfl_xor(colacc, 16, 32);                    // combine s-halves
    if (lane < 16)
        colsum[(size_t)bh * Ss + t0 + lm] = colacc;
}

// ---------------------------------------------------------------------------
// Pass 2: fused attention output.
//   w[s,t] = score[s,t] * N_C[t]^-0.5 ;  out[s,:] = (w @ V) / max(sum_t w, eps)
// One wave per (bh, 16-wide s-tile). Q fragments + 16x64 accumulators resident.
// K and V 16-row tiles arrive as ONE 3D TDM transfer (2 slices, stride 768),
// double-buffered in LDS; score tile relayouts C->A through a small LDS pad.
// ---------------------------------------------------------------------------
__global__ __launch_bounds__(32)
void attn_out_kernel(const float* __restrict__ qkv, const float* __restrict__ qn,
                     const float* __restrict__ kn, const float* __restrict__ colsum,
                     float* __restrict__ attn_out)
{
#if USE_TDM
    __shared__ float kv[2][2 * 16 * 64];   // [buf][slice*1024 + t*64 + d], 2 x 8KB
#endif
    __shared__ float wlds[16 * 18];        // pad 18 -> aligned v2f rows, no conflicts

    const int lane = threadIdx.x;
    const int half = lane >> 4;
    const int lm   = lane & 15;
    const int widx = blockIdx.x;
    const int bh = widx >> 7;
    const int st = widx & 127;
    const int b  = bh / Hh, h = bh % Hh;
    const int s0 = st << 4;

    const float* qbase = qkv + (size_t)b * Ss * TRIPLE + h * Dd;
    const float* kbase = qbase + HID;          // K rows;  V rows at +HID more
    const float* vbase = qbase + 2 * HID;

    // resident Q A-fragments (16 x 64)
    v2f qa[16];
#pragma unroll
    for (int c = 0; c < 16; ++c)
        qa[c] = *(const v2f*)(qbase + (size_t)(s0 + lm) * TRIPLE + 4 * c + 2 * half);
    const float qv = qn[(size_t)bh * Ss + s0 + lm];

    v8f acc0 = {}, acc1 = {}, acc2 = {}, acc3 = {};
    float zacc[8] = {0.f, 0.f, 0.f, 0.f, 0.f, 0.f, 0.f, 0.f};

#if USE_TDM
    const unsigned kv_off = (unsigned)(size_t)&kv[0][0];
    tdm_tile_load(kv_off, kbase, 2, (unsigned)HID);          // prologue: t0 = 0
#endif

    for (int t0 = 0; t0 < Ss; t0 += 16) {
        // ---- QK^T tile ----
        v8f dot = {};
#if USE_TDM
        const int cur = (t0 >> 4) & 1;
        TDM_WAIT();
        if (t0 + 16 < Ss)
            tdm_tile_load(kv_off + (unsigned)((cur ^ 1) * 8192),
                          kbase + (size_t)(t0 + 16) * TRIPLE, 2, (unsigned)HID);
        const float* kt = &kv[cur][0];          // K slice: [t][d]
        const float* vt = kt + 1024;            // V slice: [t][d]
#pragma unroll
        for (int c = 0; c < 16; ++c) {
            v2f kbf = *(const v2f*)(kt + lm * 64 + 4 * c + 2 * half);
            dot = WMMA_F32(qa[c], kbf, dot);
        }
#else
        if (t0 + 16 < Ss)
            __builtin_prefetch(kbase + (size_t)(t0 + 16 + lm) * TRIPLE, 0, 0);
#pragma unroll
        for (int c = 0; c < 16; ++c) {
            v2f kbf = *(const v2f*)(kbase + (size_t)(t0 + lm) * TRIPLE + 4 * c + 2 * half);
            dot = WMMA_F32(qa[c], kbf, dot);
        }
#endif
        const float knv = kn[(size_t)bh * Ss + t0 + lm];
        const float rC  = rsqrtf(colsum[(size_t)bh * Ss + t0 + lm]);  // N_C^-0.5

        // ---- weights, Z accumulation, stage tile C-layout -> LDS ----
#pragma unroll
        for (int r = 0; r < 8; ++r) {
            float qrow = __shfl(qv, r + 8 * half, 32);
            float d2 = qrow + knv - 2.0f * dot[r];
            float g  = sqrtf(fmaxf(d2, 0.0f));
            float w  = __powf(1.0f + g, PPOW) * rC;
            zacc[r] += w;
            wlds[(r + 8 * half) * 18 + lm] = w;
        }
        __syncthreads();   // single-wave WG: ordering fence for LDS relayout

        // ---- P @ V : w (16s x 16t) x V (16t x 64d) ----
#pragma unroll
        for (int c = 0; c < 4; ++c) {
            v2f wa = *(const v2f*)(&wlds[lm * 18 + 4 * c + 2 * half]);
            v2f vb;
#if USE_TDM
            const float* v0 = vt + (4 * c + 2 * half) * 64;
            const float* v1 = v0 + 64;
#else
            const float* v0 = vbase + (size_t)(t0 + 4 * c + 2 * half) * TRIPLE;
            const float* v1 = v0 + TRIPLE;
#endif
            vb.x = v0[ 0 + lm]; vb.y = v1[ 0 + lm]; acc0 = WMMA_F32(wa, vb, acc0);
            vb.x = v0[16 + lm]; vb.y = v1[16 + lm]; acc1 = WMMA_F32(wa, vb, acc1);
            vb.x = v0[32 + lm]; vb.y = v1[32 + lm]; acc2 = WMMA_F32(wa, vb, acc2);
            vb.x = v0[48 + lm]; vb.y = v1[48 + lm]; acc3 = WMMA_F32(wa, vb, acc3);
        }
        __syncthreads();
    }

    // ---- row normalizer: reduce zacc across the 16 lanes of each half ----
    float invz[8];
#pragma unroll
    for (int r = 0; r < 8; ++r) {
        float z = zacc[r];
        z += __shfl_xor(z, 1, 32);
        z += __shfl_xor(z, 2, 32);
        z += __shfl_xor(z, 4, 32);
        z += __shfl_xor(z, 8, 32);
        invz[r] = 1.0f / fmaxf(z, EPSF);
    }

    // ---- write attn_out in [B, S, H*D] (matches reference head-merge) ----
#pragma unroll
    for (int r = 0; r < 8; ++r) {
        float* orow = attn_out + (size_t)(b * Ss + s0 + r + 8 * half) * HID + h * Dd + lm;
        orow[ 0] = acc0[r] * invz[r];
        orow[16] = acc1[r] * invz[r];
        orow[32] = acc2[r] * invz[r];
        orow[48] = acc3[r] * invz[r];
    }
    (void)vbase;
}

// ---------------------------------------------------------------------------
// Host-side launcher
// ---------------------------------------------------------------------------
extern "C" void kernel_launch(void* const* d_in, const int* in_sizes, int n_in,
                              void* d_out, int out_size, void* d_ws, size_t ws_size,
                              hipStream_t stream)
{
    const float* x    = (const float*)d_in[0];
    const float* Wqkv = (const float*)d_in[1];
    const float* bqkv = (const float*)d_in[2];
    const float* Wout = (const float*)d_in[3];
    const float* bout = (const float*)d_in[4];
    float* out = (float*)d_out;

    char* ws = (char*)d_ws;
    size_t off = 0;
    auto alloc_f = [&](size_t nelem) -> float* {
        float* p = (float*)(ws + off);
        off += ((nelem * sizeof(float) + 255) / 256) * 256;
        return p;
    };
    float* qkv = alloc_f((size_t)Bb * Ss * TRIPLE);   // 75.5 MB
    float* qn  = alloc_f((size_t)Bb * Hh * Ss);
    float* kn  = alloc_f((size_t)Bb * Hh * Ss);
    float* cs  = alloc_f((size_t)Bb * Hh * Ss);
    float* ao  = alloc_f((size_t)Bb * Ss * HID);      // 25.2 MB

    // 1) QKV projection
    {
        const int M = Bb * Ss, N = TRIPLE, K = HID;
        const int tiles = (M / 16) * (N / 64);
        gemm_wmma_f32<<<(tiles + 7) / 8, 256, 0, stream>>>(x, Wqkv, bqkv, qkv, M, N, K);
    }
    // 2) row norms of q and k
    {
        const int n = Bb * Hh * Ss;
        norms_kernel<<<(n + 255) / 256, 256, 0, stream>>>(qkv, qn, kn);
    }
    // 3) column sums of the power-law kernel
    colsum_kernel<<<Bb * Hh * (Ss / 16), 32, 0, stream>>>(qkv, qn, kn, cs);
    // 4) fused normalized attention + P@V
    attn_out_kernel<<<Bb * Hh * (Ss / 16), 32, 0, stream>>>(qkv, qn, kn, cs, ao);
    // 5) output projection
    {
        const int M = Bb * Ss, N = HID, K = HID;
        const int tiles = (M / 16) * (N / 64);
        gemm_wmma_f32<<<(tiles + 7) / 8, 256, 0, stream>>>(ao, Wout, bout, out, M, N, K);
    }
    (void)in_sizes; (void)n_in; (void)out_size; (void)ws_size;
}